// I_FocalLoss_75282186764507
// MI455X (gfx1250) — compile-verified
//
#include <hip/hip_runtime.h>
#include <hip/hip_bf16.h>

// Focal loss with mislabel suppression, MI455X (gfx1250).
// HBM-bound streaming row reduction: 524 MB / 23.3 TB/s ~= 22.5 us floor.
// Wave-per-row online softmax; gfx1250 async global->LDS double buffering
// (GLOBAL_LOAD_ASYNC_TO_LDS_B128 + s_wait_asynccnt), NT-load fallback otherwise.

typedef float v4f __attribute__((ext_vector_type(4)));
typedef int   v4i __attribute__((ext_vector_type(4)));

#define NUM_C      1000
#define V4_PER_ROW 250          // NUM_C / 4
#define WAVES      8
#define TPB        256
#define TILES      8            // rows per wave
#define L2E        1.44269504088896340736f
#define LN2        0.69314718055994530942f

#if defined(__HIP_DEVICE_COMPILE__) && defined(__gfx1250__) && \
    __has_builtin(__builtin_amdgcn_global_load_async_to_lds_b128)
#define USE_ASYNC 1
#else
#define USE_ASYNC 0
#endif

#if USE_ASYNC
// pointer-to-(AS1 v4i) and pointer-to-(AS3 v4i), matching the builtin's
// "vV4i*1V4i*3IiIi" signature reported by the compiler.
typedef __attribute__((address_space(1))) v4i* as1_v4i_ptr;
typedef __attribute__((address_space(3))) v4i* as3_v4i_ptr;

#if __has_builtin(__builtin_amdgcn_s_wait_asynccnt)
#define WAIT_ASYNC(n)                                     \
    do {                                                  \
        __builtin_amdgcn_s_wait_asynccnt(n);              \
        asm volatile("" ::: "memory");                    \
    } while (0)
#else
#define WAIT_ASYNC(n) asm volatile("s_wait_asynccnt %0" ::"i"(n) : "memory")
#endif
#endif

__global__ __launch_bounds__(TPB) void focal_loss_rows(
    const float* __restrict__ x, const float* __restrict__ alpha,
    const int* __restrict__ target, float* __restrict__ partial, int nrows)
{
    const int lane      = threadIdx.x & 31;
    const int wave      = threadIdx.x >> 5;
    const int blockBase = blockIdx.x * (WAVES * TILES);

#if USE_ASYNC
    __shared__ __align__(16) float buf[2][WAVES][NUM_C];
#endif
    __shared__ float wsum[WAVES];

    float acc = 0.0f;

#if USE_ASYNC
    {   // prefetch tile 0 for this wave's row
        const int r0 = blockBase + wave;
        if (r0 < nrows) {
            const float* g = x + (size_t)r0 * NUM_C;
            float*       l = &buf[0][wave][0];
#pragma unroll
            for (int j = 0; j < 8; ++j) {
                const int idx = j * 32 + lane;
                if (idx < V4_PER_ROW) {
                    __builtin_amdgcn_global_load_async_to_lds_b128(
                        (as1_v4i_ptr)(g + idx * 4),
                        (as3_v4i_ptr)(l + idx * 4),
                        0, 0);
                }
            }
        }
    }
#endif

    for (int t = 0; t < TILES; ++t) {
        const int row = blockBase + t * WAVES + wave;

#if USE_ASYNC
        bool issuedNext = false;
        if (t + 1 < TILES) {
            const int nrow = row + WAVES;
            if (nrow < nrows) {
                const float* g = x + (size_t)nrow * NUM_C;
                float*       l = &buf[(t + 1) & 1][wave][0];
#pragma unroll
                for (int j = 0; j < 8; ++j) {
                    const int idx = j * 32 + lane;
                    if (idx < V4_PER_ROW) {
                        __builtin_amdgcn_global_load_async_to_lds_b128(
                            (as1_v4i_ptr)(g + idx * 4),
                            (as3_v4i_ptr)(l + idx * 4),
                            0, 0);
                    }
                }
                issuedNext = true;
            }
        }
        if (issuedNext) { WAIT_ASYNC(8); }  // async loads retire in order: first 8 (current tile) done
        else            { WAIT_ASYNC(0); }
#endif

        if (row < nrows) {
#if USE_ASYNC
            const float* rsrc = &buf[t & 1][wave][0];
#else
            const float* rsrc = x + (size_t)row * NUM_C;
            if (t + 1 < TILES && row + WAVES < nrows)  // stream next row toward L2
                __builtin_prefetch(x + (size_t)(row + WAVES) * NUM_C + lane * 32, 0, 0);
#endif
            // base-2 online softmax: mL = max(x)*log2e, s = sum 2^(x*log2e - mL)
            float mL   = -__builtin_inff();
            float s    = 0.0f;
            int   bidx = 0x7fffffff;
#pragma unroll
            for (int j = 0; j < 8; ++j) {
                const int idx = j * 32 + lane;
                if (idx < V4_PER_ROW) {
#if USE_ASYNC
                    v4f v = *(const v4f*)(rsrc + idx * 4);
#else
                    v4f v = __builtin_nontemporal_load((const v4f*)(rsrc + idx * 4));
#endif
#pragma unroll
                    for (int k = 0; k < 4; ++k) {
                        const float xL = v[k] * L2E;
                        if (xL > mL) {
                            s    = s * exp2f(mL - xL) + 1.0f;
                            mL   = xL;
                            bidx = idx * 4 + k;
                        } else {
                            s += exp2f(xL - mL);
                        }
                    }
                }
            }
            // wave32 butterfly reduce of (max, scaled-sum, first argmax)
#pragma unroll
            for (int off = 16; off > 0; off >>= 1) {
                const float om = __shfl_xor(mL, off, 32);
                const float os = __shfl_xor(s, off, 32);
                const int   oi = __shfl_xor(bidx, off, 32);
                if (om > mL) {
                    s    = s * exp2f(mL - om) + os;
                    mL   = om;
                    bidx = oi;
                } else if (om < mL) {
                    s += os * exp2f(om - mL);
                } else {
                    s += os;
                    bidx = (oi < bidx) ? oi : bidx;
                }
            }
            const int tg = target[row];
#if USE_ASYNC
            const float xt = rsrc[tg];
#else
            const float xt = x[(size_t)row * NUM_C + tg];
#endif
            const float lse      = LN2 * (mL + log2f(s));
            const float logpt    = xt - lse;
            const float pt       = exp2f(logpt * L2E);
            const float at       = alpha[tg];
            const float omp      = 1.0f - pt;
            const float fl       = -omp * omp * logpt * at;
            const float ptop     = 1.0f / s;   // max term contributes 2^0 = 1
            const bool  suppress = (ptop > 0.5f) && (bidx != tg);
            acc += suppress ? 1.0e-5f : fl;
        }
    }

    if (lane == 0) wsum[wave] = acc;   // acc is wave-uniform after butterfly
    __syncthreads();
    if (threadIdx.x == 0) {
        float b = 0.0f;
#pragma unroll
        for (int w = 0; w < WAVES; ++w) b += wsum[w];
        partial[blockIdx.x] = b;
    }
}

__global__ __launch_bounds__(256) void focal_loss_reduce(
    const float* __restrict__ partial, int nparts, float invN,
    float* __restrict__ out)
{
    __shared__ float sm[256];
    float a = 0.0f;
    for (int i = threadIdx.x; i < nparts; i += 256) a += partial[i];
    sm[threadIdx.x] = a;
    __syncthreads();
#pragma unroll
    for (int off = 128; off > 0; off >>= 1) {
        if ((int)threadIdx.x < off) sm[threadIdx.x] += sm[threadIdx.x + off];
        __syncthreads();
    }
    if (threadIdx.x == 0) out[0] = sm[0] * invN;
}

extern "C" void kernel_launch(void* const* d_in, const int* in_sizes, int n_in,
                              void* d_out, int out_size, void* d_ws, size_t ws_size,
                              hipStream_t stream)
{
    const float* x      = (const float*)d_in[0];
    const float* alpha  = (const float*)d_in[1];
    const int*   target = (const int*)d_in[2];
    float*       out    = (float*)d_out;

    const int nrows        = in_sizes[2];           // N from target length
    const int rowsPerBlock = WAVES * TILES;         // 64
    const int blocks       = (nrows + rowsPerBlock - 1) / rowsPerBlock;
    float*    partial      = (float*)d_ws;          // blocks * 4 bytes

    focal_loss_rows<<<blocks, TPB, 0, stream>>>(x, alpha, target, partial, nrows);
    focal_loss_reduce<<<1, 256, 0, stream>>>(partial, blocks, 1.0f / (float)nrows, out);
}